// ProposalLayer_31679678775602
// MI455X (gfx1250) — compile-verified
//
#include <hip/hip_runtime.h>
#include <hip/hip_bf16.h>
#include <stdint.h>

// ---- problem constants (fixed shapes from the reference) ----
#define BATCH    16
#define NA       9
#define FH       128
#define FW       128
#define FHW      (FH*FW)        // 16384
#define NPI      (FHW*NA)       // 147456 anchors per image
#define PRE_NMS  2000
#define POST_NMS 300
#define NBINS    65536          // histogram over top-16 bits of sortable score
#define CAND_CAP 4096           // candidate cap per image (top-2000 + bin overshoot)
#define PRE_PAD  2048

#ifndef __has_builtin
#define __has_builtin(x) 0
#endif
#if __has_builtin(__builtin_amdgcn_global_load_async_to_lds_b128) && \
    __has_builtin(__builtin_amdgcn_s_wait_asynccnt)
#define USE_ASYNC 1
#else
#define USE_ASYNC 0
#endif

#if USE_ASYNC
#define AS_GLOBAL __attribute__((address_space(1)))
#define AS_LDS    __attribute__((address_space(3)))
// exact parameter element type from the builtin signature (vV4i*1 V4i*3 Ii Ii)
typedef int v4i __attribute__((vector_size(4 * sizeof(int))));
// global -> LDS async copy (per-lane addressed), tracked by ASYNCcnt
__device__ __forceinline__ void async_gather_b128(void* lds_dst, const void* gsrc) {
  __builtin_amdgcn_global_load_async_to_lds_b128(
      (AS_GLOBAL v4i*)(gsrc), (AS_LDS v4i*)(lds_dst), 0, 0);
}
#endif

// monotonic float -> u32 mapping (order-preserving incl. -inf)
__device__ __forceinline__ unsigned sortable_f32(float f) {
  unsigned u = __float_as_uint(f);
  return (u & 0x80000000u) ? ~u : (u | 0x80000000u);
}

// ---------------- bitonic sort + async LDS staging + greedy NMS + output ----------------
// (defined first so the device-asm snippet shows the async-to-LDS path)
__global__ __launch_bounds__(1024) void k_sort_nms_out(const unsigned long long* __restrict__ cand,
                                                       const unsigned* __restrict__ cnts,
                                                       const float* __restrict__ proposals,
                                                       const float* __restrict__ thr_p,
                                                       float* __restrict__ out) {
  const int b = blockIdx.x, tid = threadIdx.x;
  __shared__ unsigned long long keys[CAND_CAP];   // 32KB, reused as boxes after extraction
  __shared__ int           idxs[PRE_PAD];         // 8KB
  __shared__ unsigned char keep[PRE_PAD];         // 2KB
  __shared__ short         srcRank[POST_NMS];
  float4* boxes = reinterpret_cast<float4*>(keys); // PRE_NMS*16B <= CAND_CAP*8B

  unsigned cnt = cnts[b]; if (cnt > CAND_CAP) cnt = CAND_CAP;

  // stage candidate keys into LDS: contiguous async global->LDS copy (2 keys per b128)
#if USE_ASYNC
  for (int e2 = tid * 2; e2 < CAND_CAP; e2 += 2048)
    async_gather_b128(&keys[e2], &cand[(size_t)b * CAND_CAP + e2]);
  __builtin_amdgcn_s_wait_asynccnt(0);
  __syncthreads();
  // zero the tail past cnt (stale ws bytes loaded above are overwritten before any use)
  for (int e = tid; e < CAND_CAP; e += 1024)
    if (e >= (int)cnt) keys[e] = 0ull;
#else
  for (int e = tid; e < CAND_CAP; e += 1024)
    keys[e] = (e < (int)cnt) ? cand[(size_t)b * CAND_CAP + e] : 0ull;
#endif

  // bitonic sort, descending (first loop iteration begins with a barrier)
  for (unsigned k = 2; k <= CAND_CAP; k <<= 1) {
    for (unsigned j = k >> 1; j > 0; j >>= 1) {
      __syncthreads();
      for (unsigned e = tid; e < CAND_CAP; e += 1024) {
        unsigned ixj = e ^ j;
        if (ixj > e) {
          unsigned long long va = keys[e], vb = keys[ixj];
          bool desc = ((e & k) == 0);
          if (desc ? (va < vb) : (va > vb)) { keys[e] = vb; keys[ixj] = va; }
        }
      }
    }
  }
  __syncthreads();

  // extract top PRE_NMS: index + liveness (dead = padding or masked -inf score)
  for (int r = tid; r < PRE_PAD; r += 1024) {
    if (r < PRE_NMS) {
      unsigned long long k = keys[r];
      unsigned scu = (unsigned)(k >> 32);
      bool live = (scu >= 0x80000000u);          // any fg prob >= +0.0
      int id = live ? (int)(0xFFFFFFFFu - (unsigned)k) : 0;
      if (id < 0 || id >= NPI) id = 0;
      idxs[r] = id; keep[r] = live ? 1 : 0;
    } else { idxs[r] = 0; keep[r] = 0; }
  }
  __syncthreads();   // all key reads done; safe to overwrite with boxes

  // gather top boxes into LDS (per-lane addressed async copy)
  for (int r = tid; r < PRE_NMS; r += 1024) {
    const float* src = proposals + ((size_t)b * NPI + (size_t)idxs[r]) * 4;
#if USE_ASYNC
    async_gather_b128(&boxes[r], src);
#else
    boxes[r] = *(const float4*)src;
#endif
  }
#if USE_ASYNC
  __builtin_amdgcn_s_wait_asynccnt(0);
#endif
  __syncthreads();

  // greedy NMS: identical recurrence to the reference fori_loop
  const float th = thr_p[0];
  for (int i = 0; i < PRE_NMS; ++i) {
    __syncthreads();
    if (!keep[i]) continue;               // uniform: same LDS value for all threads
    float4 bi = boxes[i];
    float area_i = (bi.z - bi.x + 1.0f) * (bi.w - bi.y + 1.0f);
    for (int j = i + 1 + tid; j < PRE_NMS; j += 1024) {
      if (!keep[j]) continue;
      float4 bj = boxes[j];
      float iw = fminf(bi.z, bj.z) - fmaxf(bi.x, bj.x) + 1.0f;
      float ih = fminf(bi.w, bj.w) - fmaxf(bi.y, bj.y) + 1.0f;
      iw = fmaxf(iw, 0.0f); ih = fmaxf(ih, 0.0f);
      float inter = iw * ih;
      float area_j = (bj.z - bj.x + 1.0f) * (bj.w - bj.y + 1.0f);
      if (inter / (area_i + area_j - inter) > th) keep[j] = 0;
    }
  }
  __syncthreads();

  for (int k = tid; k < POST_NMS; k += 1024) srcRank[k] = -1;
  __syncthreads();
  if (tid == 0) {
    int c = 0;
    for (int i = 0; i < PRE_NMS && c < POST_NMS; ++i)
      if (keep[i]) { srcRank[c] = (short)i; ++c; }
  }
  __syncthreads();

  // output row: [b, x1, y1, x2, y2]; zero-filled past the kept count (ref scatters into zeros)
  for (int k = tid; k < POST_NMS; k += 1024) {
    float* row = out + ((size_t)b * POST_NMS + k) * 5;
    row[0] = (float)b;
    int sr = srcRank[k];
    if (sr >= 0) {
      float4 bx = boxes[sr];
      row[1] = bx.x; row[2] = bx.y; row[3] = bx.z; row[4] = bx.w;
    } else {
      row[1] = 0.0f; row[2] = 0.0f; row[3] = 0.0f; row[4] = 0.0f;
    }
  }
}

// ---------------- zero histogram + counters ----------------
__global__ void k_zero_u32(unsigned* p, int n) {
  int i = blockIdx.x * blockDim.x + threadIdx.x;
  if (i < n) p[i] = 0u;
}

// ---------------- softmax + decode + clip + mask + histogram ----------------
__global__ void k_decode(const float* __restrict__ scores,
                         const float* __restrict__ deltas,
                         const float* __restrict__ anchors,
                         const float* __restrict__ im_info,
                         float* __restrict__ proposals,      // [BATCH, NPI, 4]
                         unsigned* __restrict__ sc,          // [BATCH*NPI] sortable masked score
                         unsigned* __restrict__ hist) {      // [BATCH, NBINS]
  long long g = (long long)blockIdx.x * blockDim.x + threadIdx.x;
  if (g >= (long long)BATCH * NPI) return;
  int b = (int)(g / NPI);
  int r = (int)(g % NPI);
  int a  = r % NA;
  int hw = r / NA;             // h*FW + w

  // fg prob: softmax over channel pair (a, a+NA) at (h,w)
  size_t sbase = ((size_t)b * (2*NA)) * FHW + hw;
  float s0 = scores[sbase + (size_t)a        * FHW];
  float s1 = scores[sbase + (size_t)(a + NA) * FHW];
  float fg = 1.0f / (1.0f + expf(s0 - s1));

  // deltas: channel = a*4 + c
  size_t dbase = ((size_t)b * (4*NA) + (size_t)a * 4) * FHW + hw;
  float dx = deltas[dbase];
  float dy = deltas[dbase + (size_t)FHW];
  float dw = deltas[dbase + (size_t)2*FHW];
  float dh = deltas[dbase + (size_t)3*FHW];

  float4 an = ((const float4*)anchors)[r];
  float wdt = an.z - an.x + 1.0f, hgt = an.w - an.y + 1.0f;
  float cx  = an.x + 0.5f * wdt, cy  = an.y + 0.5f * hgt;
  float pcx = dx * wdt + cx,     pcy = dy * hgt + cy;
  float pw  = expf(dw) * wdt,    ph  = expf(dh) * hgt;
  float x1 = pcx - 0.5f * pw, y1 = pcy - 0.5f * ph;
  float x2 = pcx + 0.5f * pw, y2 = pcy + 0.5f * ph;

  float wmax = im_info[b*4 + 1] - 1.0f;
  float hmax = im_info[b*4 + 0] - 1.0f;
  x1 = fminf(fmaxf(x1, 0.0f), wmax);  x2 = fminf(fmaxf(x2, 0.0f), wmax);
  y1 = fminf(fmaxf(y1, 0.0f), hmax);  y2 = fminf(fmaxf(y2, 0.0f), hmax);

  ((float4*)proposals)[g] = make_float4(x1, y1, x2, y2);

  bool valid = ((x2 - x1 + 1.0f) >= 16.0f * im_info[b*4 + 3]) &&
               ((y2 - y1 + 1.0f) >= 16.0f * im_info[b*4 + 2]);
  unsigned u = sortable_f32(valid ? fg : -INFINITY);
  sc[g] = u;
  atomicAdd(&hist[(size_t)b * NBINS + (u >> 16)], 1u);
}

// ---------------- per-image threshold bin (suffix scan of histogram) ----------------
__global__ __launch_bounds__(1024) void k_find_thresh(const unsigned* __restrict__ hist,
                                                      unsigned* __restrict__ tbin) {
  const int b = blockIdx.x, tid = threadIdx.x;
  const unsigned* h = hist + (size_t)b * NBINS;
  __shared__ unsigned csum[1024];
  __shared__ unsigned suf[1024];
  const int base = tid * (NBINS / 1024);   // 64 bins per thread
  unsigned s = 0;
  for (int i = 0; i < NBINS/1024; ++i) s += h[base + i];
  csum[tid] = s;
  __syncthreads();
  if (tid == 0) {
    unsigned run = 0;
    for (int q = 1023; q >= 0; --q) { suf[q] = run; run += csum[q]; }
  }
  __syncthreads();
  unsigned above = suf[tid];
  if (above < PRE_NMS && above + csum[tid] >= PRE_NMS) {
    unsigned run = above;
    for (int i = NBINS/1024 - 1; i >= 0; --i) {
      run += h[base + i];
      if (run >= PRE_NMS) { tbin[b] = (unsigned)(base + i); break; }
    }
  }
}

// ---------------- compact candidate keys ----------------
__global__ void k_compact(const unsigned* __restrict__ sc,
                          const unsigned* __restrict__ tbin,
                          unsigned* __restrict__ cnts,
                          unsigned long long* __restrict__ cand) {
  long long g = (long long)blockIdx.x * blockDim.x + threadIdx.x;
  if (g >= (long long)BATCH * NPI) return;
  int b = (int)(g / NPI);
  int r = (int)(g % NPI);
  unsigned s = sc[g];
  if ((s >> 16) >= tbin[b]) {
    unsigned p = atomicAdd(&cnts[b], 1u);
    if (p < CAND_CAP) {
      // key = (score<<32) | (~index): distinct keys -> deterministic total order,
      // ties in score broken by smaller index first (matches lax.top_k)
      cand[(size_t)b * CAND_CAP + p] =
          ((unsigned long long)s << 32) | (unsigned long long)(0xFFFFFFFFu - (unsigned)r);
    }
  }
}

// ---------------- launcher ----------------
extern "C" void kernel_launch(void* const* d_in, const int* in_sizes, int n_in,
                              void* d_out, int out_size, void* d_ws, size_t ws_size,
                              hipStream_t stream) {
  (void)in_sizes; (void)n_in; (void)out_size;
  const float* scores    = (const float*)d_in[0];
  const float* deltas    = (const float*)d_in[1];
  const float* anchors   = (const float*)d_in[2];
  const float* im_info   = (const float*)d_in[3];
  // d_in[4..6] = num_anchors/pre_nms/post_nms (int, fixed), d_in[7] = nms_thresh (float)
  const float* nms_thresh = (const float*)d_in[7];

  float* out       = (float*)d_out;                                  // [BATCH,POST_NMS,5]
  float* proposals = out + (size_t)BATCH * POST_NMS * 5;             // [BATCH,NPI,4]

  // workspace layout
  const size_t SC_BYTES   = (size_t)BATCH * NPI   * sizeof(unsigned);      // 9,437,184
  const size_t HIST_BYTES = (size_t)BATCH * NBINS * sizeof(unsigned);      // 4,194,304
  const size_t CTR_BYTES  = 128;                                           // cnts[16]+tbin[16]
  const size_t CAND_BYTES = (size_t)BATCH * CAND_CAP * sizeof(unsigned long long);
  if (ws_size < SC_BYTES + HIST_BYTES + CTR_BYTES + CAND_BYTES) return;    // defensive

  char* ws = (char*)d_ws;
  unsigned* sc   = (unsigned*)ws;
  unsigned* hist = (unsigned*)(ws + SC_BYTES);
  unsigned* cnts = hist + (size_t)BATCH * NBINS;
  unsigned* tbin = cnts + 16;
  unsigned long long* cand = (unsigned long long*)(ws + SC_BYTES + HIST_BYTES + CTR_BYTES);

  const int nzero = BATCH * NBINS + 32;
  k_zero_u32<<<(nzero + 255) / 256, 256, 0, stream>>>(hist, nzero);

  const long long ntot = (long long)BATCH * NPI;
  const int blks = (int)((ntot + 255) / 256);
  k_decode<<<blks, 256, 0, stream>>>(scores, deltas, anchors, im_info, proposals, sc, hist);
  k_find_thresh<<<BATCH, 1024, 0, stream>>>(hist, tbin);
  k_compact<<<blks, 256, 0, stream>>>(sc, tbin, cnts, cand);
  k_sort_nms_out<<<BATCH, 1024, 0, stream>>>(cand, cnts, proposals, nms_thresh, out);
}